// LIFLayer_64819646431519
// MI455X (gfx1250) — compile-verified
//
#include <hip/hip_runtime.h>
#include <math.h>

typedef __attribute__((ext_vector_type(2))) float v2f;
typedef __attribute__((ext_vector_type(8))) float v8f;
typedef __attribute__((ext_vector_type(4))) unsigned int u32x4;
typedef __attribute__((ext_vector_type(8))) int i32x8;
typedef __attribute__((ext_vector_type(4))) int i32x4;

#define B_SZ 32
#define T_SZ 1024
#define DIN  512
#define H_SZ 512
#define HT   64          // h-columns per workgroup
#define KPRS 160         // floats per K-pair row of the weight slice (64 float2 pairs + pad; 160%64=32 -> halves on disjoint banks)
#define ISTR 516         // effective staged-input row stride produced by TDM padding (512 + 2 pads of 2 dwords)
#define CSTR 68          // currents slab stride (8*68%64=32 -> lane halves disjoint)

// ---- Tensor Data Mover: DMA one 16 x 512 f32 input chunk into LDS with padding ----
// D# group0: count=1, lds_addr, global_addr, type=2
// D# group1: data_size=4B, pad_enable, pad_interval=256 dwords, pad_amount=2 dwords,
//            tensor_dim0=512, tensor_dim1=16, tile_dim0=512, tile_dim1=16, dim0_stride=512
__device__ __forceinline__ void tdm_load_chunk(const float* gsrc, unsigned lds_addr)
{
    unsigned long long ga = (unsigned long long)(size_t)gsrc;
    u32x4 g0;
    g0[0] = 1u;                                                   // count=1 (user descriptor)
    g0[1] = lds_addr;                                             // LDS byte address
    g0[2] = (unsigned)ga;                                         // global_addr[31:0]
    g0[3] = (unsigned)((ga >> 32) & 0x01FFFFFFu) | (2u << 30);    // global_addr[56:32] | type=2

    i32x8 g1;
    g1[0] = (int)((2u << 16) | (1u << 20) | (7u << 22) | (1u << 25)); // data_size=4B | pad_en | interval=256dw | pad=2dw
    g1[1] = (int)(512u << 16);        // tensor_dim0 = 512 (bits[79:48], low 16 here)
    g1[2] = (int)(16u  << 16);        // tensor_dim1 = 16
    g1[3] = (int)(512u << 16);        // tile_dim0 = 512
    g1[4] = 16;                       // tile_dim1 = 16
    g1[5] = 512;                      // tensor_dim0_stride = 512
    g1[6] = 0;
    g1[7] = 0;

    i32x4 z4 = {0, 0, 0, 0};
#if defined(__clang_major__) && (__clang_major__ >= 23)
    i32x8 z8 = {0, 0, 0, 0, 0, 0, 0, 0};
    __builtin_amdgcn_tensor_load_to_lds(g0, g1, z4, z4, z8, 0);
#else
    __builtin_amdgcn_tensor_load_to_lds(g0, g1, z4, z4, 0);
#endif
}

__global__ __launch_bounds__(64) void lif_fused(const float* __restrict__ inp,
                                                const float* __restrict__ ker,
                                                const float* __restrict__ bias,
                                                float* __restrict__ out_spk,
                                                float* __restrict__ out_vm,
                                                float* __restrict__ out_vf,
                                                float* __restrict__ out_rate)
{
    __shared__ float kds[256 * KPRS];       // weight slice, K-pair interleaved: kds[p][2h..2h+1] = {W[2p][h], W[2p+1][h]}
    __shared__ float ilds[2][16 * ISTR];    // double-buffered staged inputs (filled by TDM)
    __shared__ float cur[16 * CSTR];        // currents slab
    __shared__ float red[64];

    const int tid  = threadIdx.x;
    const int lane = tid & 31;
    const int wv   = tid >> 5;          // wave 0..1
    const int ln16 = lane & 15;
    const int lhi  = lane >> 4;         // lane half (0/1)
    const int kh   = lhi * 2;           // K sub-offset in a K=4 group (ISA A/B layout)
    const int h0   = blockIdx.x * HT;
    const int b    = blockIdx.y;

    // ---- load weight slice into K-pair-interleaved LDS layout (once) ----
    {
        const float4* ks = (const float4*)ker;
        for (int i = tid; i < 256 * 16; i += 64) {
            int p  = i >> 4;            // K-pair index
            int hq = i & 15;            // float4 group within 64 h
            float4 ra = ks[(size_t)(2 * p)     * (H_SZ / 4) + (h0 >> 2) + hq];
            float4 rb = ks[(size_t)(2 * p + 1) * (H_SZ / 4) + (h0 >> 2) + hq];
            float* dst = &kds[p * KPRS + 8 * hq];
            ((float4*)dst)[0] = make_float4(ra.x, rb.x, ra.y, rb.y);
            ((float4*)dst)[1] = make_float4(ra.z, rb.z, ra.w, rb.w);
        }
    }

    const int  hoff0 = wv * 32;
    const int  hoff1 = wv * 32 + 16;
    const float bv0  = bias[h0 + hoff0 + ln16];
    const float bv1  = bias[h0 + hoff1 + ln16];

    const unsigned lds0 = (unsigned)(size_t)&ilds[0][0];   // flat low-32 bits == LDS byte offset
    const unsigned lds1 = (unsigned)(size_t)&ilds[1][0];
    const float* gbase  = inp + (size_t)b * T_SZ * DIN;

    if (wv == 0) tdm_load_chunk(gbase, lds0);              // prologue: chunk 0 -> buf 0

    const float asyn = 0.818730753f;    // exp(-1/5)
    const float amem = 0.951229424f;    // exp(-1/20)
    const float bmem = 0.048770575f;    // 1 - exp(-1/20)

    float vmem = 0.0f, isyn = 0.0f, ssum = 0.0f;

    for (int t0 = 0; t0 < T_SZ; t0 += 16) {
        const int cb = (t0 >> 4) & 1;

        if (wv == 0) __builtin_amdgcn_s_wait_tensorcnt(0); // current buffer landed
        __syncthreads();                                    // publish buffer; prior scan done with cur

        if (wv == 0 && (t0 + 16) < T_SZ)                    // kick DMA for next chunk into the other buffer
            tdm_load_chunk(gbase + (size_t)(t0 + 16) * DIN, cb ? lds0 : lds1);
        asm volatile("" ::: "memory");

        // ---- fp32 WMMA: 2 tiles/wave, K split in two 256-halves (TDM pad shifts k>=256 by +2) ----
        v8f acc0, acc1;
#pragma unroll
        for (int i = 0; i < 8; ++i) { acc0[i] = bv0; acc1[i] = bv1; }

        const float* ibuf = &ilds[cb][0];
#pragma unroll
        for (int hseg = 0; hseg < 2; ++hseg) {
            const float* abase = ibuf + ln16 * ISTR + hseg * 258 + kh;            // A row m=ln16
            const float* bb0   = &kds[(hseg * 128 + lhi) * KPRS + 2 * (hoff0 + ln16)];
            const float* bb1   = &kds[(hseg * 128 + lhi) * KPRS + 2 * (hoff1 + ln16)];
#pragma unroll 8
            for (int kk = 0; kk < 256; kk += 4) {
                v2f a  = *(const v2f*)(abase + kk);        // ds_load_b64, conflict-free
                v2f b0 = *(const v2f*)(bb0 + kk * 80);     // one aligned b64 per fragment: no v_mov glue
                v2f b1 = *(const v2f*)(bb1 + kk * 80);
                acc0 = __builtin_amdgcn_wmma_f32_16x16x4_f32(false, a, false, b0, (short)0, acc0, false, false);
                acc1 = __builtin_amdgcn_wmma_f32_16x16x4_f32(false, a, false, b1, (short)0, acc1, false, false);
            }
        }

        // C layout: VGPR v -> row v (lanes 0-15) / row v+8 (lanes 16-31)
        const int rbase = lhi * 8;
#pragma unroll
        for (int v = 0; v < 8; ++v) {
            cur[(rbase + v) * CSTR + hoff0 + ln16] = acc0[v];
            cur[(rbase + v) * CSTR + hoff1 + ln16] = acc1[v];
        }
        __syncthreads();

        // ---- LIF scan: thread tid owns column h0+tid, 16 sequential steps ----
        size_t obase = ((size_t)b * T_SZ + t0) * H_SZ + h0 + tid;
#pragma unroll 4
        for (int r = 0; r < 16; ++r) {
            float c = cur[r * CSTR + tid];
            isyn = asyn * isyn + c;
            vmem = amem * vmem + bmem * isyn;
            float spk  = (vmem >= 1.0f) ? 1.0f : 0.0f;
            float surr = 2.0f / (1.0f + coshf(4.0f * (vmem - 1.0f)));  // beta/(2(1+cosh)), beta=4
            float sw   = 2.0f * spk - surr;       // spikes + stopgrad(spikes - surrogate), forward value
            vmem = (spk > 0.0f) ? 0.0f : vmem;    // reset
            out_spk[obase] = sw;
            out_vm[obase]  = vmem;                // post-reset membrane
            ssum += sw;
            obase += H_SZ;
        }
    }

    // v_final (post-reset carry at t = T-1)
    out_vf[(size_t)b * H_SZ + h0 + tid] = vmem;

    // spike_rate: block reduce + one atomic per workgroup (pre-scaled by 1/(B*T*H) = 2^-24)
    red[tid] = ssum;
    __syncthreads();
    if (tid == 0) {
        float s = 0.0f;
#pragma unroll
        for (int i = 0; i < 64; ++i) s += red[i];
        atomicAdd(out_rate, s * 5.9604644775390625e-8f);
    }
}

__global__ void zero_rate(float* p)
{
    if (threadIdx.x == 0 && blockIdx.x == 0) *p = 0.0f;
}

extern "C" void kernel_launch(void* const* d_in, const int* in_sizes, int n_in,
                              void* d_out, int out_size, void* d_ws, size_t ws_size,
                              hipStream_t stream)
{
    const float* inp  = (const float*)d_in[0];   // [32,1024,512] f32
    const float* ker  = (const float*)d_in[1];   // [512,512]     f32
    const float* bias = (const float*)d_in[2];   // [512]         f32

    float* out = (float*)d_out;
    const size_t nBTH = (size_t)B_SZ * T_SZ * H_SZ;
    float* out_spk  = out;                       // [B,T,H]
    float* out_vm   = out + nBTH;                // [B,T,H]
    float* out_vf   = out + 2 * nBTH;            // [B,H]
    float* out_rate = out + 2 * nBTH + (size_t)B_SZ * H_SZ;  // scalar

    hipLaunchKernelGGL(zero_rate, dim3(1), dim3(1), 0, stream, out_rate);
    hipLaunchKernelGGL(lif_fused, dim3(H_SZ / HT, B_SZ), dim3(64), 0, stream,
                       inp, ker, bias, out_spk, out_vm, out_vf, out_rate);
}